// BatchSpanCrossEntropyLoss_31868657336514
// MI455X (gfx1250) — compile-verified
//
#include <hip/hip_runtime.h>
#include <math.h>

typedef float v2f __attribute__((ext_vector_type(2)));
typedef float v8f __attribute__((ext_vector_type(8)));

#define BB 256
#define SS 16384
#define MASK_WPS 512                 // SS/32 words per sample
#define MASK_TOT (BB * MASK_WPS)     // 131072 words per mask

// ---- workspace layout -------------------------------------------------------
// u32  [0        , 131072)   beg_mask
// u32  [131072   , 262144)   end_mask
// f32  F[0    ..  256)       num_ann[b]
// f32  F[256  ..  768)       maxps [b][2]   per-sample running max
// f32  F[768  .. 1280)       sums  [b][2]   sum exp w.r.t. own max
// f32  F[1280 .. 1792)       numer [b][2]   masked sum exp w.r.t. own max
// f32  F[1792 .. 5888)       rhs   [256][16] packed GEMM RHS
// f32  F[5888 .. 9984)       gout  [256][16] GEMM output
// total ~1.09 MB

__global__ void k_zero(unsigned int* w, int n) {
    int i = blockIdx.x * blockDim.x + threadIdx.x;
    if (i < n) w[i] = 0u;
}

// Build begin/end bitmasks (dedup == multi-hot clamp) and per-sample label sum.
__global__ void k_scatter(const int* __restrict__ begins,
                          const int* __restrict__ ends,
                          const int* __restrict__ labels,
                          unsigned int* __restrict__ beg_mask,
                          unsigned int* __restrict__ end_mask,
                          float* __restrict__ num_ann) {
    __shared__ int red[256];
    int b = blockIdx.x >> 6;                 // 64 blocks per sample
    int s = ((blockIdx.x & 63) << 8) + threadIdx.x;
    int idx = b * SS + s;
    int lab = labels[idx];
    if (lab > 0) {
        int bg = begins[idx] & (SS - 1);
        int ed = ends[idx] & (SS - 1);
        atomicOr(&beg_mask[b * MASK_WPS + (bg >> 5)], 1u << (bg & 31));
        atomicOr(&end_mask[b * MASK_WPS + (ed >> 5)], 1u << (ed & 31));
    }
    red[threadIdx.x] = lab;
    __syncthreads();
    for (int off = 128; off > 0; off >>= 1) {
        if (threadIdx.x < off) red[threadIdx.x] += red[threadIdx.x + off];
        __syncthreads();
    }
    if (threadIdx.x == 0) atomicAdd(&num_ann[b], (float)red[0]);
}

// One pass over logits: online softmax stats (max, sum) + bit-masked numerators.
__global__ void k_stats(const float* __restrict__ logits,
                        const unsigned int* __restrict__ beg_mask,
                        const unsigned int* __restrict__ end_mask,
                        float* __restrict__ maxps,
                        float* __restrict__ sums,
                        float* __restrict__ numer) {
    __shared__ float sm0[256], ss0[256], sn0[256];
    __shared__ float sm1[256], ss1[256], sn1[256];
    int b = blockIdx.x;
    int tid = threadIdx.x;
    const float* lg = logits + (size_t)b * SS * 2;
    const unsigned int* bm = beg_mask + b * MASK_WPS;
    const unsigned int* em = end_mask + b * MASK_WPS;

    float m0 = -INFINITY, s0 = 0.f, n0 = 0.f;
    float m1 = -INFINITY, s1 = 0.f, n1 = 0.f;
    for (int i = 0; i < SS / 256; ++i) {
        int s = i * 256 + tid;
        float2 v = *(const float2*)(lg + 2 * s);
        unsigned int bit = s & 31;
        float hb = (bm[s >> 5] >> bit) & 1u ? 1.f : 0.f;
        float he = (em[s >> 5] >> bit) & 1u ? 1.f : 0.f;
        // channel 0 (begin)
        float nm = fmaxf(m0, v.x);
        float r  = __expf(m0 - nm);
        float e  = __expf(v.x - nm);
        s0 = s0 * r + e;  n0 = n0 * r + hb * e;  m0 = nm;
        // channel 1 (end)
        nm = fmaxf(m1, v.y);
        r  = __expf(m1 - nm);
        e  = __expf(v.y - nm);
        s1 = s1 * r + e;  n1 = n1 * r + he * e;  m1 = nm;
    }
    sm0[tid] = m0; ss0[tid] = s0; sn0[tid] = n0;
    sm1[tid] = m1; ss1[tid] = s1; sn1[tid] = n1;
    __syncthreads();
    for (int off = 128; off > 0; off >>= 1) {
        if (tid < off) {
            float ma = sm0[tid], mb = sm0[tid + off];
            float M = fmaxf(ma, mb), ra = __expf(ma - M), rb = __expf(mb - M);
            sm0[tid] = M;
            ss0[tid] = ss0[tid] * ra + ss0[tid + off] * rb;
            sn0[tid] = sn0[tid] * ra + sn0[tid + off] * rb;
            ma = sm1[tid]; mb = sm1[tid + off];
            M = fmaxf(ma, mb); ra = __expf(ma - M); rb = __expf(mb - M);
            sm1[tid] = M;
            ss1[tid] = ss1[tid] * ra + ss1[tid + off] * rb;
            sn1[tid] = sn1[tid] * ra + sn1[tid + off] * rb;
        }
        __syncthreads();
    }
    if (tid == 0) {
        maxps[2 * b + 0] = sm0[0]; sums[2 * b + 0] = ss0[0]; numer[2 * b + 0] = sn0[0];
        maxps[2 * b + 1] = sm1[0]; sums[2 * b + 1] = ss1[0]; numer[2 * b + 1] = sn1[0];
    }
}

// doc-max across same-block samples; rescale per-sample stats; pack GEMM RHS.
__global__ void k_docmax_rhs(const int* __restrict__ block_ids,
                             const float* __restrict__ maxps,
                             const float* __restrict__ sums,
                             const float* __restrict__ numer,
                             const float* __restrict__ num_ann,
                             float* __restrict__ rhs) {
    __shared__ int bid[BB];
    int b = threadIdx.x;
    bid[b] = block_ids[b];
    __syncthreads();
    int my = bid[b];
    float dm0 = -INFINITY, dm1 = -INFINITY;
    for (int j = 0; j < BB; ++j) {
        if (bid[j] == my) {
            dm0 = fmaxf(dm0, maxps[2 * j + 0]);
            dm1 = fmaxf(dm1, maxps[2 * j + 1]);
        }
    }
    float r0 = __expf(maxps[2 * b + 0] - dm0);
    float r1 = __expf(maxps[2 * b + 1] - dm1);
    float* row = rhs + b * 16;
    row[0] = sums[2 * b + 0] * r0;   // -> denom0 after eq@
    row[1] = sums[2 * b + 1] * r1;   // -> denom1
    row[2] = numer[2 * b + 0] * r0;  // -> agg numer0
    row[3] = numer[2 * b + 1] * r1;  // -> agg numer1
    row[4] = num_ann[b];             // -> num_per_doc
    row[5] = 1.0f;                   // -> row count sum(eq)
    for (int c = 6; c < 16; ++c) row[c] = 0.0f;
}

// gout[256][16] = eq[256][256] @ rhs[256][16] via V_WMMA_F32_16X16X4_F32.
// 1 block, 512 threads = 16 waves; wave w computes rows 16w..16w+15.
__global__ void __launch_bounds__(512)
k_eq_gemm(const int* __restrict__ block_ids,
          const float* __restrict__ rhs,
          float* __restrict__ gout) {
    __shared__ int bid[BB];
    if (threadIdx.x < BB) bid[threadIdx.x] = block_ids[threadIdx.x];
    __syncthreads();

    int lane = threadIdx.x & 31;       // wave32
    int wave = threadIdx.x >> 5;
    int m0 = wave * 16;
    int half = lane >> 4;              // 0: K+0/K+1 lanes, 1: K+2/K+3 lanes
    int l = lane & 15;

    int mrow = m0 + l;
    int mybid = bid[mrow];

    v8f acc = {};
    for (int k0 = 0; k0 < BB; k0 += 4) {
        int ka = k0 + 2 * half;
        v2f a, bf;
        a.x = (mybid == bid[ka + 0]) ? 1.0f : 0.0f;   // A: eq[m][k], 16x4 layout
        a.y = (mybid == bid[ka + 1]) ? 1.0f : 0.0f;
        bf.x = rhs[(ka + 0) * 16 + l];                // B: rhs[k][n], 4x16 layout
        bf.y = rhs[(ka + 1) * 16 + l];
        acc = __builtin_amdgcn_wmma_f32_16x16x4_f32(
            false, a, false, bf, (short)0, acc, false, false);
    }
    // D layout: VGPR i, lanes 0-15 -> M=i,N=lane; lanes 16-31 -> M=i+8,N=lane-16
    for (int i = 0; i < 8; ++i) {
        int row = m0 + i + 8 * half;
        gout[row * 16 + l] = acc[i];
    }
}

// Final scalar loss.
__global__ void k_finalize(const float* __restrict__ gout, float* __restrict__ out) {
    __shared__ float red[BB];
    int b = threadIdx.x;
    const float* g = gout + b * 16;
    float d0 = g[0], d1 = g[1], a0 = g[2], a1 = g[3], npd = g[4], cnt = g[5];
    float wm = (npd > 0.0f) ? 1.0f : 0.0f;
    float wo = 1.0f - wm;
    float lc0 = __logf(a0 / d0 + wo) * wm;
    float lc1 = __logf(a1 / d1 + wo) * wm;
    red[b] = -(lc0 + lc1) / cnt;
    __syncthreads();
    for (int off = 128; off > 0; off >>= 1) {
        if (b < off) red[b] += red[b + off];
        __syncthreads();
    }
    if (b == 0) out[0] = red[0];
}

extern "C" void kernel_launch(void* const* d_in, const int* in_sizes, int n_in,
                              void* d_out, int out_size, void* d_ws, size_t ws_size,
                              hipStream_t stream) {
    const float* logits = (const float*)d_in[0];
    const int* begins   = (const int*)d_in[1];
    const int* ends     = (const int*)d_in[2];
    const int* labels   = (const int*)d_in[3];
    const int* bids     = (const int*)d_in[4];

    unsigned int* beg_mask = (unsigned int*)d_ws;
    unsigned int* end_mask = beg_mask + MASK_TOT;
    float* F       = (float*)(end_mask + MASK_TOT);
    float* num_ann = F;
    float* maxps   = F + 256;
    float* sums    = F + 768;
    float* numer   = F + 1280;
    float* rhs     = F + 1792;
    float* gout    = F + 5888;

    // zero masks + num_ann (contiguous: 2*MASK_TOT u32 words + 256 floats)
    int nz = 2 * MASK_TOT + 256;
    k_zero<<<(nz + 255) / 256, 256, 0, stream>>>(beg_mask, nz);

    k_scatter<<<BB * (SS / 256), 256, 0, stream>>>(begins, ends, labels,
                                                   beg_mask, end_mask, num_ann);

    k_stats<<<BB, 256, 0, stream>>>(logits, beg_mask, end_mask, maxps, sums, numer);

    k_docmax_rhs<<<1, 256, 0, stream>>>(bids, maxps, sums, numer, num_ann, rhs);

    k_eq_gemm<<<1, 512, 0, stream>>>(bids, rhs, gout);

    k_finalize<<<1, 256, 0, stream>>>(gout, (float*)d_out);
}